// DAGNN_6760278524489
// MI455X (gfx1250) — compile-verified
//
#include <hip/hip_runtime.h>
#include <hip/hip_bf16.h>

#define N_NODES 50000
#define D_FEAT  128
#define K_HOPS  10
#define EPB     64   // edges staged per block in the scatter kernel

typedef __attribute__((ext_vector_type(2))) float v2f;
typedef __attribute__((ext_vector_type(8))) float v8f;

// -------- softmax over the 11 attention logits (single thread; trivial) ------
__global__ void dagnn_softmax_kernel(const float* __restrict__ att,
                                     float* __restrict__ w, int n) {
    if (threadIdx.x == 0 && blockIdx.x == 0) {
        float m = att[0];
        for (int i = 1; i < n; ++i) m = fmaxf(m, att[i]);
        float e[16];
        float s = 0.f;
        for (int i = 0; i < n; ++i) { e[i] = __expf(att[i] - m); s += e[i]; }
        float inv = 1.0f / s;
        for (int i = 0; i < n; ++i) w[i] = e[i] * inv;
    }
}

// -------- init: out = w[0]*x, zero first scatter destination -----------------
__global__ __launch_bounds__(256) void dagnn_init_kernel(
    const float* __restrict__ x, float* __restrict__ out,
    float* __restrict__ hzero, const float* __restrict__ w) {
    size_t i = (size_t)blockIdx.x * blockDim.x + threadIdx.x;  // float4 index
    float w0 = w[0];
    float4 xv = ((const float4*)x)[i];
    float4 o; o.x = w0 * xv.x; o.y = w0 * xv.y; o.z = w0 * xv.z; o.w = w0 * xv.w;
    ((float4*)out)[i] = o;
    float4 z; z.x = 0.f; z.y = 0.f; z.z = 0.f; z.w = 0.f;
    ((float4*)hzero)[i] = z;
}

// -------- f32 atomic add (lowers to no-return global_atomic_add_f32) ---------
__device__ __forceinline__ void atomic_add_f32(float* p, float v) {
    __hip_atomic_fetch_add(p, v, __ATOMIC_RELAXED, __HIP_MEMORY_SCOPE_AGENT);
}

// -------- scatter hop: h_dst[dst] += h_src[src] over all edges ---------------
// One wave per edge-row-copy: lane l moves float4 at column 4*l (512B/row,
// fully coalesced). Edge (src,dst) pairs are staged into LDS with CDNA5
// async global->LDS loads, published with s_wait_asynccnt + barrier.
__global__ __launch_bounds__(256) void dagnn_scatter_kernel(
    const float* __restrict__ hsrc, float* __restrict__ hdst,
    const int* __restrict__ esrc, const int* __restrict__ edst, int nE) {
    __shared__ int ls[EPB];
    __shared__ int ld[EPB];
    const int t  = threadIdx.x;
    const int e0 = blockIdx.x * EPB;

    // threads 0..63 stage src indices, 64..127 stage dst indices
    if (t < 2 * EPB) {
        const int  i     = t & (EPB - 1);
        const bool isDst = (t >= EPB);
        const int  e     = e0 + i;
        if (e < nE) {
            const int* gp = (isDst ? edst : esrc) + e;
            int*       lp = (isDst ? ld : ls) + i;
            unsigned           lds_addr = (unsigned)(size_t)lp;            // LDS byte offset
            unsigned long long gaddr    = (unsigned long long)(size_t)gp;  // 64-bit VGPR pair
            asm volatile("global_load_async_to_lds_b32 %0, %1, off"
                         :: "v"(lds_addr), "v"(gaddr) : "memory");
        }
    }
    asm volatile("s_wait_asynccnt 0" ::: "memory");
    __syncthreads();

    const int wave = t >> 5;
    const int lane = t & 31;
    const int col  = lane * 4;
#pragma unroll
    for (int j = 0; j < EPB / 8; ++j) {      // 8 waves x 8 edges = 64 edges
        const int ei = wave * (EPB / 8) + j;
        const int e  = e0 + ei;
        if (e < nE) {
            const int s = ls[ei];
            const int d = ld[ei];
            const float4 v = *(const float4*)(hsrc + (size_t)s * D_FEAT + col);
            float* dp = hdst + (size_t)d * D_FEAT + col;
            atomic_add_f32(dp + 0, v.x);
            atomic_add_f32(dp + 1, v.y);
            atomic_add_f32(dp + 2, v.z);
            atomic_add_f32(dp + 3, v.w);
        }
    }
}

// -------- WMMA combine: out_tile += w[k] * h_tile, and zero next buffer ------
// Exact f32 math on the matrix pipe: per 16x16 tile, 4 chained
// V_WMMA_F32_16X16X4_F32 ops with selection matrices
//   A_g[m][kk] = w  iff (m>>2)==g && (m&3)==kk, else 0
//   B_g[kk][n] = H[node0 + 4g + kk][f0 + n]
// so chained D = C + w * H_tile. Grid is exact (3125*8 tiles), EXEC all-ones.
__global__ __launch_bounds__(256) void dagnn_combine_wmma_kernel(
    float* __restrict__ out, const float* __restrict__ h,
    float* __restrict__ hz, const float* __restrict__ wv, int k) {
    const float    w      = wv[k];
    const unsigned lane   = threadIdx.x & 31u;
    const unsigned waveId = blockIdx.x * (blockDim.x >> 5) + (threadIdx.x >> 5);
    const unsigned node0  = (waveId >> 3) * 16u;   // 3125 node tiles
    const unsigned f0     = (waveId & 7u) * 16u;   // 8 feature tiles
    const unsigned n      = lane & 15u;
    const unsigned hi     = lane >> 4;             // 0: K/M rows 0..; 1: upper half

    // C/D layout: VGPR i -> (M = i + 8*hi, N = n)
    v8f c;
#pragma unroll
    for (int i = 0; i < 8; ++i)
        c[i] = out[(size_t)(node0 + i + hi * 8u) * D_FEAT + f0 + n];

    const unsigned mA = lane & 15u;                // A rows: lanes 0-15 and 16-31 both M=0..15
    const unsigned k0 = hi ? 2u : 0u;              // A/B VGPR0 holds K=0 (lo) / K=2 (hi)
    const unsigned k1 = hi ? 3u : 1u;              // A/B VGPR1 holds K=1 (lo) / K=3 (hi)
#pragma unroll
    for (unsigned g = 0; g < 4; ++g) {
        v2f a, b;
        b.x = h[(size_t)(node0 + 4u * g + k0) * D_FEAT + f0 + n];
        b.y = h[(size_t)(node0 + 4u * g + k1) * D_FEAT + f0 + n];
        a.x = ((mA >> 2) == g && (mA & 3u) == k0) ? w : 0.0f;
        a.y = ((mA >> 2) == g && (mA & 3u) == k1) ? w : 0.0f;
        c = __builtin_amdgcn_wmma_f32_16x16x4_f32(false, a, false, b,
                                                  (short)0, c, false, false);
    }

#pragma unroll
    for (int i = 0; i < 8; ++i) {
        const size_t off = (size_t)(node0 + i + hi * 8u) * D_FEAT + f0 + n;
        out[off] = c[i];
        hz[off]  = 0.0f;    // zero the buffer that will be the next scatter target
    }
}

extern "C" void kernel_launch(void* const* d_in, const int* in_sizes, int n_in,
                              void* d_out, int out_size, void* d_ws, size_t ws_size,
                              hipStream_t stream) {
    (void)n_in; (void)out_size; (void)ws_size;
    const float* x   = (const float*)d_in[0];
    const float* att = (const float*)d_in[1];
    const int*   ei  = (const int*)d_in[2];      // harness passes integer inputs as int32
    const int    nE  = in_sizes[2] / 2;          // [2, E] -> E edges
    float*       out = (float*)d_out;

    // workspace: [w(64B pad to 256B)] [hA: N*D f32] [hB: N*D f32]
    float* w  = (float*)d_ws;
    float* hA = (float*)((char*)d_ws + 256);
    float* hB = hA + (size_t)N_NODES * D_FEAT;

    const int* esrc = ei;         // edge_index[0] : message sources
    const int* edst = ei + nE;    // edge_index[1] : scatter destinations

    dagnn_softmax_kernel<<<1, 32, 0, stream>>>(att, w, K_HOPS + 1);

    const int initBlocks = (N_NODES * D_FEAT / 4) / 256;             // 6250
    dagnn_init_kernel<<<initBlocks, 256, 0, stream>>>(x, out, hA, w);

    const int scatterBlocks = (nE + EPB - 1) / EPB;                  // 9766
    const int combineBlocks = (N_NODES / 16) * (D_FEAT / 16) / 8;    // 3125

    const float* src = x;
    float*       dst = hA;
    for (int k = 1; k <= K_HOPS; ++k) {
        float* other = (dst == hA) ? hB : hA;
        dagnn_scatter_kernel<<<scatterBlocks, 256, 0, stream>>>(src, dst, esrc, edst, nE);
        dagnn_combine_wmma_kernel<<<combineBlocks, 256, 0, stream>>>(out, dst, other, w, k);
        src = dst;
        dst = other;
    }
}